// GNNPolicyNetLite_22737556865376
// MI455X (gfx1250) — compile-verified
//
#include <hip/hip_runtime.h>
#include <math.h>

// Problem constants (from reference)
#define NN 100000     // nodes
#define EE 3200000    // edges
#define FIN 32        // input feature dim
#define HD 64         // hidden dim
#define NL 4          // layers
#define ASZ 6158      // action space
#define GG 64         // graphs

typedef __attribute__((ext_vector_type(2))) float v2f;
typedef __attribute__((ext_vector_type(8))) float v8f;

// ---------------------------------------------------------------------------
// Utility: zero a float buffer (count must be multiple of 4)
// ---------------------------------------------------------------------------
__global__ void zero_kernel(float* __restrict__ p, int n4) {
    int i = blockIdx.x * blockDim.x + threadIdx.x;
    if (i < n4) ((float4*)p)[i] = make_float4(0.f, 0.f, 0.f, 0.f);
}

// ---------------------------------------------------------------------------
// In-degree via atomics, then inv_deg = 1/max(deg,1) in place
// ---------------------------------------------------------------------------
__global__ void degree_kernel(const int* __restrict__ dst, float* __restrict__ deg) {
    int e = blockIdx.x * blockDim.x + threadIdx.x;
    if (e < EE) atomicAdd(&deg[dst[e]], 1.0f);
}

__global__ void invdeg_kernel(float* __restrict__ deg) {
    int i = blockIdx.x * blockDim.x + threadIdx.x;
    if (i < NN) deg[i] = 1.0f / fmaxf(deg[i], 1.0f);
}

// ---------------------------------------------------------------------------
// Encoder: h = relu(x @ W_enc^T + b_enc)   [NN,32] x [64,32]^T -> [NN,64]
// One block = 128 threads = 4 waves; block -> 16-node tile, wave -> 16-col tile.
// f32 WMMA 16x16x4, K loop over 32.
// ---------------------------------------------------------------------------
__global__ __launch_bounds__(128) void encoder_kernel(
    const float* __restrict__ x, const float* __restrict__ W,
    const float* __restrict__ b, float* __restrict__ h) {
    int lane = threadIdx.x & 31;
    int wave = threadIdx.x >> 5;
    int l15  = lane & 15;
    int hi   = lane >> 4;        // 0 | 1
    int koff = hi << 1;          // 0 | 2
    int m0   = blockIdx.x << 4;
    int row  = m0 + l15;
    int col  = (wave << 4) + l15;

    const float* xr = x + row * FIN + koff;
    const float* wr = W + col * FIN + koff;

    v8f c = {0.f, 0.f, 0.f, 0.f, 0.f, 0.f, 0.f, 0.f};
#pragma unroll
    for (int k = 0; k < FIN; k += 4) {
        v2f a  = *(const v2f*)(xr + k);
        v2f bb = *(const v2f*)(wr + k);
        c = __builtin_amdgcn_wmma_f32_16x16x4_f32(false, a, false, bb,
                                                  (short)0, c, false, false);
    }
    float bias  = b[col];
    int   mbase = m0 + (hi << 3);
#pragma unroll
    for (int v = 0; v < 8; ++v) {
        float val = c[v] + bias;
        h[(mbase + v) * HD + col] = fmaxf(val, 0.0f);
    }
}

// ---------------------------------------------------------------------------
// Edge aggregation: agg[dst] += h[src]   (mean applied later via inv_deg)
// 16 lanes per edge, each handles a float4 feature chunk -> coalesced gather,
// 4 scalar f32 atomics per lane. h (25.6MB) lives in L2.
// ---------------------------------------------------------------------------
__global__ void agg_kernel(const int* __restrict__ src, const int* __restrict__ dst,
                           const float* __restrict__ h, float* __restrict__ agg) {
    int t = blockIdx.x * blockDim.x + threadIdx.x;   // < EE*16 = 51.2M
    int e = t >> 4;
    if (e >= EE) return;
    int f = (t & 15) << 2;
    int s = src[e];
    int d = dst[e];
    float4 v = *(const float4*)(h + s * HD + f);
    float* ap = agg + d * HD + f;
    atomicAdd(ap + 0, v.x);
    atomicAdd(ap + 1, v.y);
    atomicAdd(ap + 2, v.z);
    atomicAdd(ap + 3, v.w);
}

// ---------------------------------------------------------------------------
// SAGE layer: hout = relu( (agg*inv_deg) @ Wl^T + bl + hin @ Wr^T ) + hin
// Same tiling as encoder; two K=64 passes accumulate into one f32 WMMA C tile.
// ---------------------------------------------------------------------------
__global__ __launch_bounds__(128) void sage_kernel(
    const float* __restrict__ agg, const float* __restrict__ hin,
    const float* __restrict__ invd,
    const float* __restrict__ Wl, const float* __restrict__ bl,
    const float* __restrict__ Wr, float* __restrict__ hout) {
    int lane = threadIdx.x & 31;
    int wave = threadIdx.x >> 5;
    int l15  = lane & 15;
    int hi   = lane >> 4;
    int koff = hi << 1;
    int m0   = blockIdx.x << 4;
    int row  = m0 + l15;
    int col  = (wave << 4) + l15;

    float s = invd[row];
    const float* ar = agg + row * HD + koff;
    const float* hr = hin + row * HD + koff;
    const float* wl = Wl + col * HD + koff;
    const float* wr = Wr + col * HD + koff;

    v8f c = {0.f, 0.f, 0.f, 0.f, 0.f, 0.f, 0.f, 0.f};
#pragma unroll
    for (int k = 0; k < HD; k += 4) {   // mean-aggregated neighbors through Wl
        v2f a = *(const v2f*)(ar + k);
        a.x *= s; a.y *= s;
        v2f bb = *(const v2f*)(wl + k);
        c = __builtin_amdgcn_wmma_f32_16x16x4_f32(false, a, false, bb,
                                                  (short)0, c, false, false);
    }
#pragma unroll
    for (int k = 0; k < HD; k += 4) {   // root features through Wr
        v2f a  = *(const v2f*)(hr + k);
        v2f bb = *(const v2f*)(wr + k);
        c = __builtin_amdgcn_wmma_f32_16x16x4_f32(false, a, false, bb,
                                                  (short)0, c, false, false);
    }
    float bias  = bl[col];
    int   mbase = m0 + (hi << 3);
#pragma unroll
    for (int v = 0; v < 8; ++v) {
        int   m    = mbase + v;
        float prev = hin[m * HD + col];
        hout[m * HD + col] = fmaxf(c[v] + bias, 0.0f) + prev;
    }
}

// ---------------------------------------------------------------------------
// Global mean pool: atomic sums + counts, then normalize in place
// ---------------------------------------------------------------------------
__global__ void pool_sum_kernel(const float* __restrict__ h, const int* __restrict__ batch,
                                float* __restrict__ gsum) {
    int t = blockIdx.x * blockDim.x + threadIdx.x;   // < NN*HD
    if (t >= NN * HD) return;
    int node = t >> 6;
    int f    = t & 63;
    atomicAdd(&gsum[batch[node] * HD + f], h[t]);
}

__global__ void pool_cnt_kernel(const int* __restrict__ batch, float* __restrict__ gcnt) {
    int i = blockIdx.x * blockDim.x + threadIdx.x;
    if (i < NN) atomicAdd(&gcnt[batch[i]], 1.0f);
}

__global__ void pool_norm_kernel(float* __restrict__ gsum, const float* __restrict__ gcnt) {
    int t = blockIdx.x * blockDim.x + threadIdx.x;
    if (t >= GG * HD) return;
    gsum[t] = gsum[t] / fmaxf(gcnt[t >> 6], 1.0f);
}

// ---------------------------------------------------------------------------
// Policy head: out = graph_repr @ Wp^T + bp   [64,64] x [6158,64]^T
// Block = 128 threads = 4 waves covering all 64 rows; block per 16-col tile.
// N tail (6158 % 16 = 14) masked on B loads and stores.
// ---------------------------------------------------------------------------
__global__ __launch_bounds__(128) void policy_kernel(
    const float* __restrict__ gr, const float* __restrict__ Wp,
    const float* __restrict__ bp, float* __restrict__ out) {
    int lane = threadIdx.x & 31;
    int wave = threadIdx.x >> 5;
    int l15  = lane & 15;
    int hi   = lane >> 4;
    int koff = hi << 1;
    int row  = (wave << 4) + l15;        // graph id 0..63
    int n    = (blockIdx.x << 4) + l15;  // action column
    bool nv  = (n < ASZ);

    const float* ar = gr + row * HD + koff;
    const float* wr = Wp + n * HD + koff;

    v2f zz; zz.x = 0.f; zz.y = 0.f;
    v8f c = {0.f, 0.f, 0.f, 0.f, 0.f, 0.f, 0.f, 0.f};
#pragma unroll
    for (int k = 0; k < HD; k += 4) {
        v2f a  = *(const v2f*)(ar + k);
        v2f bb = nv ? *(const v2f*)(wr + k) : zz;
        c = __builtin_amdgcn_wmma_f32_16x16x4_f32(false, a, false, bb,
                                                  (short)0, c, false, false);
    }
    if (nv) {
        float bias  = bp[n];
        int   gbase = (wave << 4) + (hi << 3);
#pragma unroll
        for (int v = 0; v < 8; ++v) out[(gbase + v) * ASZ + n] = c[v] + bias;
    }
}

// ---------------------------------------------------------------------------
// Value head: tanh(graph_repr @ Wv^T + bv)  -> GG scalars
// ---------------------------------------------------------------------------
__global__ void value_kernel(const float* __restrict__ gr, const float* __restrict__ Wv,
                             const float* __restrict__ bv, float* __restrict__ out) {
    int g = blockIdx.x * blockDim.x + threadIdx.x;
    if (g >= GG) return;
    float acc = 0.f;
#pragma unroll 8
    for (int k = 0; k < HD; ++k) acc += gr[g * HD + k] * Wv[k];
    out[g] = tanhf(acc + bv[0]);
}

// ---------------------------------------------------------------------------
// Launch
// ---------------------------------------------------------------------------
extern "C" void kernel_launch(void* const* d_in, const int* in_sizes, int n_in,
                              void* d_out, int out_size, void* d_ws, size_t ws_size,
                              hipStream_t stream) {
    const float* x     = (const float*)d_in[0];
    const int*   ei    = (const int*)d_in[1];     // [2,E]: row0=src, row1=dst
    const int*   batch = (const int*)d_in[2];
    const float* W_enc = (const float*)d_in[3];
    const float* b_enc = (const float*)d_in[4];
    const float* Wl    = (const float*)d_in[5];   // [L,H,H]
    const float* bl    = (const float*)d_in[6];   // [L,H]
    const float* Wr    = (const float*)d_in[7];   // [L,H,H]
    const float* Wp    = (const float*)d_in[8];   // [A,H]
    const float* bp    = (const float*)d_in[9];   // [A]
    const float* Wv    = (const float*)d_in[10];  // [1,H]
    const float* bv    = (const float*)d_in[11];  // [1]
    float* out_policy  = (float*)d_out;           // [G*A]
    float* out_value   = (float*)d_out + GG * ASZ;

    const int* src = ei;
    const int* dst = ei + EE;

    // Workspace layout (floats)
    float* ws   = (float*)d_ws;
    float* deg  = ws;                    // NN      (becomes inv_deg in place)
    float* hA   = deg + NN;              // NN*HD
    float* hB   = hA + (size_t)NN * HD;  // NN*HD
    float* agg  = hB + (size_t)NN * HD;  // NN*HD
    float* gsum = agg + (size_t)NN * HD; // GG*HD
    float* gcnt = gsum + GG * HD;        // GG   (contiguous with gsum)

    // --- degree / inverse degree ---
    zero_kernel<<<(NN / 4 + 255) / 256, 256, 0, stream>>>(deg, NN / 4);
    degree_kernel<<<(EE + 255) / 256, 256, 0, stream>>>(dst, deg);
    invdeg_kernel<<<(NN + 255) / 256, 256, 0, stream>>>(deg);

    // --- encoder ---
    encoder_kernel<<<NN / 16, 128, 0, stream>>>(x, W_enc, b_enc, hA);

    // --- SAGE layers ---
    float* hc = hA;
    float* hn = hB;
    for (int i = 0; i < NL; ++i) {
        zero_kernel<<<((NN * HD / 4) + 255) / 256, 256, 0, stream>>>(agg, NN * HD / 4);
        agg_kernel<<<(EE * 16) / 256, 256, 0, stream>>>(src, dst, hc, agg);
        sage_kernel<<<NN / 16, 128, 0, stream>>>(agg, hc, deg,
                                                 Wl + (size_t)i * HD * HD, bl + i * HD,
                                                 Wr + (size_t)i * HD * HD, hn);
        float* tmp = hc; hc = hn; hn = tmp;
    }

    // --- global mean pool ---
    zero_kernel<<<((GG * HD + GG) / 4 + 255) / 256, 256, 0, stream>>>(gsum, (GG * HD + GG) / 4);
    pool_sum_kernel<<<(NN * HD) / 256, 256, 0, stream>>>(hc, batch, gsum);
    pool_cnt_kernel<<<(NN + 255) / 256, 256, 0, stream>>>(batch, gcnt);
    pool_norm_kernel<<<(GG * HD) / 256, 256, 0, stream>>>(gsum, gcnt);

    // --- heads ---
    policy_kernel<<<(ASZ + 15) / 16, 128, 0, stream>>>(gsum, Wp, bp, out_policy);
    value_kernel<<<1, GG, 0, stream>>>(gsum, Wv, bv, out_value);
}